// PFSAPhonotacticsModel_24773371364108
// MI455X (gfx1250) — compile-verified
//
#include <hip/hip_runtime.h>
#include <hip/hip_bf16.h>
#include <math.h>

#define Q 512
#define S 64
#define BATCH 1024
#define T 16
#define TILES_MAX 128   // sum_s ceil(c_s/16) <= 64 + 1024/16 = 128
#define QPAD 516        // LDS row stride in floats: 516 % 64 == 4 -> conflict-free A-frag loads

typedef __attribute__((ext_vector_type(2))) float v2f;
typedef __attribute__((ext_vector_type(8))) float v8f;

// ---------------- Kernel 1: per-state normalizer Z[q] ----------------
// Z[q] = sum_{s,p} exp(A[q,s,p]) + exp(final[q]);  invZ = 1/Z; fn = exp(final)/Z
__global__ void fsa_norm_z(const float* __restrict__ A, const float* __restrict__ fin,
                           float* __restrict__ invZ, float* __restrict__ fn) {
  const int q = blockIdx.x;
  const float* row = A + (size_t)q * (S * Q);
  float sum = 0.f;
  for (int i = threadIdx.x; i < S * Q; i += 256) sum += __expf(row[i]);
  __shared__ float red[256];
  red[threadIdx.x] = sum;
  __syncthreads();
  for (int w = 128; w > 0; w >>= 1) {
    if (threadIdx.x < w) red[threadIdx.x] += red[threadIdx.x + w];
    __syncthreads();
  }
  if (threadIdx.x == 0) {
    float fe = __expf(fin[q]);
    float zi = 1.f / (red[0] + fe);
    invZ[q] = zi;
    fn[q]   = fe * zi;
  }
}

// ---------------- Kernel 2: W[s][p][q] = exp(A[q,s,p]) * invZ[q] ----------------
// 32x32 (q,p) tile transpose through LDS: reads coalesced over p, writes coalesced over q.
__global__ void fsa_build_w(const float* __restrict__ A, const float* __restrict__ invZ,
                            float* __restrict__ W) {
  const int s  = blockIdx.z;
  const int q0 = blockIdx.x * 32;
  const int p0 = blockIdx.y * 32;
  __shared__ float tile[32][33];
  const int tx = threadIdx.x & 31;
  const int ty = threadIdx.x >> 5;  // 0..7
#pragma unroll
  for (int i = 0; i < 4; ++i) {
    int qq = q0 + ty + 8 * i;
    tile[ty + 8 * i][tx] = __expf(A[((size_t)qq * S + s) * Q + p0 + tx]) * invZ[qq];
  }
  __syncthreads();
#pragma unroll
  for (int i = 0; i < 4; ++i) {
    int pp = p0 + ty + 8 * i;
    W[((size_t)s * Q + pp) * Q + q0 + tx] = tile[tx][ty + 8 * i];
  }
}

// ---------------- Kernel 3: per-timestep symbol grouping (counting sort, 16-padded) ----------------
__global__ void fsa_groups(const int* __restrict__ xss, int* __restrict__ idxbuf,
                           int* __restrict__ symbuf) {
  const int t = blockIdx.x;
  __shared__ int hist[S];
  __shared__ int off[S];
  __shared__ int cur[S];
  const int tid = threadIdx.x;
  if (tid < S) hist[tid] = 0;
  __syncthreads();
  for (int b = tid; b < BATCH; b += 256) atomicAdd(&hist[xss[b * T + t]], 1);
  __syncthreads();
  if (tid == 0) {
    int o = 0, tileI = 0;
    for (int s = 0; s < S; ++s) {
      off[s] = o;
      cur[s] = 0;
      int nt = (hist[s] + 15) >> 4;
      for (int j = 0; j < nt; ++j) symbuf[t * TILES_MAX + tileI++] = s;
      o += nt << 4;
    }
    for (; tileI < TILES_MAX; ++tileI) symbuf[t * TILES_MAX + tileI] = -1;
  }
  for (int j = tid; j < TILES_MAX * 16; j += 256) idxbuf[t * TILES_MAX * 16 + j] = -1;
  __syncthreads();
  for (int b = tid; b < BATCH; b += 256) {
    int s = xss[b * T + t];
    int r = atomicAdd(&cur[s], 1);
    idxbuf[t * TILES_MAX * 16 + off[s] + r] = b;
  }
}

// ---------------- Kernel 4: v0[b][q] = fn[q] ----------------
__global__ void fsa_init_v(const float* __restrict__ fn, float* __restrict__ Vc) {
  int i = blockIdx.x * 256 + threadIdx.x;
  Vc[i] = fn[i & (Q - 1)];
}

// ---------------- Kernel 5: one time step, grouped WMMA GEMM ----------------
// Block = 128 threads (4 waves). Each block: one 16-row (gathered batch) M-tile x 64 N-cols.
// Each wave owns one 16x16 output tile; K=512 swept with V_WMMA_F32_16X16X4_F32.
__global__ void fsa_step(const float* __restrict__ W, const float* __restrict__ Vin,
                         float* __restrict__ Vout, const int* __restrict__ idx_t,
                         const int* __restrict__ sym_t) {
  const int tile = blockIdx.x;
  const int s = sym_t[tile];
  if (s < 0) return;  // uniform per block: inactive padding tile

  __shared__ float Vs[16][QPAD];
  __shared__ int rows[16];
  const int tid = threadIdx.x;
  if (tid < 16) rows[tid] = idx_t[tile * 16 + tid];
  __syncthreads();

  // Stage the 16 gathered V rows (16 x 512 f32) into LDS. 8 threads per row, float4.
  {
    const int r = tid >> 3;
    const int l8 = tid & 7;
    const int row = rows[r];
    const float4* src = (const float4*)(Vin + (size_t)(row < 0 ? 0 : row) * Q);
#pragma unroll
    for (int j = l8; j < Q / 4; j += 8) {
      float4 v = (row >= 0) ? src[j] : make_float4(0.f, 0.f, 0.f, 0.f);
      *(float4*)&Vs[r][4 * j] = v;
    }
  }
  __syncthreads();

  const int wave = tid >> 5;
  const int lane = tid & 31;
  const int n  = lane & 15;   // N (and M for A-frag)
  const int hi = lane >> 4;   // K half select
  const int q0 = blockIdx.y * 64 + wave * 16;

  v8f acc = {};
  const float* Wc = W + (size_t)s * Q * Q + q0 + n;  // element (p): Wc[p*Q]
#pragma unroll 4
  for (int kk = 0; kk < Q; kk += 4) {
    const int k0 = kk + 2 * hi;
    v2f a, b;
    // A-frag (16x4): lanes 0-15 M=lane K={kk,kk+1}; lanes 16-31 M=lane-16 K={kk+2,kk+3}
    a.x = Vs[n][k0];
    a.y = Vs[n][k0 + 1];
    // B-frag (4x16): lanes 0-15 hold K={kk,kk+1}, lanes 16-31 K={kk+2,kk+3}, N=lane&15
    b.x = Wc[(size_t)k0 * Q];
    b.y = Wc[(size_t)(k0 + 1) * Q];
    acc = __builtin_amdgcn_wmma_f32_16x16x4_f32(false, a, false, b, (short)0, acc,
                                                false, false);
  }

  // D layout: VGPR r holds row M = r + 8*hi, col N = lane&15. Scatter to Vout rows.
#pragma unroll
  for (int r = 0; r < 8; ++r) {
    const int m = r + 8 * hi;
    const int row = rows[m];
    if (row >= 0) Vout[(size_t)row * Q + q0 + n] = acc[r];
  }
}

// ---------------- Kernel 6: out[b] = log( init . v_T[b] ) ----------------
__global__ void fsa_final(const float* __restrict__ V, const float* __restrict__ init,
                          float* __restrict__ out) {
  const int b = blockIdx.x;
  float sum = 0.f;
  for (int q = threadIdx.x; q < Q; q += 128) sum += init[q] * V[(size_t)b * Q + q];
  __shared__ float red[128];
  red[threadIdx.x] = sum;
  __syncthreads();
  for (int w = 64; w > 0; w >>= 1) {
    if (threadIdx.x < w) red[threadIdx.x] += red[threadIdx.x + w];
    __syncthreads();
  }
  if (threadIdx.x == 0) out[b] = logf(red[0]);
}

extern "C" void kernel_launch(void* const* d_in, const int* in_sizes, int n_in,
                              void* d_out, int out_size, void* d_ws, size_t ws_size,
                              hipStream_t stream) {
  const int*   xss  = (const int*)d_in[0];     // [B,T]
  const float* A    = (const float*)d_in[1];   // [Q,S,Q]
  const float* fin  = (const float*)d_in[2];   // [Q]
  const float* init = (const float*)d_in[3];   // [Q]
  float* out = (float*)d_out;                  // [B]

  char* ws = (char*)d_ws;
  size_t off = 0;
  auto alloc = [&](size_t bytes) -> void* {
    void* p = ws + off;
    off += (bytes + 255) & ~(size_t)255;
    return p;
  };
  float* W    = (float*)alloc((size_t)S * Q * Q * sizeof(float));     // 64 MB
  float* Va   = (float*)alloc((size_t)BATCH * Q * sizeof(float));     // 2 MB
  float* Vb   = (float*)alloc((size_t)BATCH * Q * sizeof(float));     // 2 MB
  float* invZ = (float*)alloc(Q * sizeof(float));
  float* fn   = (float*)alloc(Q * sizeof(float));
  int*   idxb = (int*)alloc((size_t)T * TILES_MAX * 16 * sizeof(int));
  int*   symb = (int*)alloc((size_t)T * TILES_MAX * sizeof(int));

  fsa_norm_z<<<Q, 256, 0, stream>>>(A, fin, invZ, fn);
  fsa_build_w<<<dim3(Q / 32, Q / 32, S), 256, 0, stream>>>(A, invZ, W);
  fsa_groups<<<T, 256, 0, stream>>>(xss, idxb, symb);
  fsa_init_v<<<(BATCH * Q) / 256, 256, 0, stream>>>(fn, Va);

  float* cur = Va;
  float* nxt = Vb;
  for (int t = T - 1; t >= 0; --t) {
    fsa_step<<<dim3(TILES_MAX, Q / 64), 128, 0, stream>>>(
        W, cur, nxt, idxb + (size_t)t * TILES_MAX * 16, symb + (size_t)t * TILES_MAX);
    float* tmp = cur; cur = nxt; nxt = tmp;
  }
  fsa_final<<<BATCH, 128, 0, stream>>>(cur, init, out);
}